// CombinedModel_15977278341388
// MI455X (gfx1250) — compile-verified
//
#include <hip/hip_runtime.h>
#include <hip/hip_bf16.h>

typedef __attribute__((ext_vector_type(2))) float v2f;
typedef __attribute__((ext_vector_type(8))) float v8f;

#define B_SZ 8
#define C_SZ 64
#define KDIM 65536          // F_WR2
#define DCLS 24576          // 3*64*16*8
#define NEGV -1e30f

// ---------------- Stage 1: split-K partials of xf(8x65536) @ wr2_w1(65536x100) ----------
// 64 blocks, each owns 1024 K values; thread n (<100) owns output column n.
__global__ void k_gemm1_partial(const float* __restrict__ x3,
                                const float* __restrict__ w1,
                                float* __restrict__ part) {
  int n = threadIdx.x;
  int blk = blockIdx.x;
  if (n >= 100) return;
  int k0 = blk * 1024;
  float acc[B_SZ] = {};
  for (int k = 0; k < 1024; ++k) {
    float w = w1[(size_t)(k0 + k) * 100 + n];   // coalesced over n
#pragma unroll
    for (int b = 0; b < B_SZ; ++b)
      acc[b] += x3[(size_t)b * KDIM + k0 + k] * w;  // broadcast load
  }
  for (int b = 0; b < B_SZ; ++b)
    part[blk * 800 + b * 100 + n] = acc[b];
}

// ---------------- Stage 2: reduce partials, two small GEMMs, postfix, clip -> boxNew(8x4)
__global__ void k_box(const float* __restrict__ part,
                      const float* __restrict__ b1,
                      const float* __restrict__ w2, const float* __restrict__ b2v,
                      const float* __restrict__ w3, const float* __restrict__ b3v,
                      float* __restrict__ boxOut) {
  __shared__ float t1[B_SZ][100];
  __shared__ float t2[B_SZ][100];
  __shared__ float bl[B_SZ][4];
  int n = threadIdx.x;
  if (n < 100) {
    for (int b = 0; b < B_SZ; ++b) {
      float s = b1[n];
      for (int blk = 0; blk < 64; ++blk) s += part[blk * 800 + b * 100 + n];
      t1[b][n] = s;
    }
  }
  __syncthreads();
  if (n < 100) {
    for (int b = 0; b < B_SZ; ++b) {
      float s = b2v[n];
      for (int m = 0; m < 100; ++m) s += t1[b][m] * w2[m * 100 + n];
      t2[b][n] = s;
    }
  }
  __syncthreads();
  if (n < 32) {
    int b = n >> 2, j = n & 3;
    float s = b3v[j];
    for (int m = 0; m < 100; ++m) s += t2[b][m] * w3[m * 4 + j];
    bl[b][j] = s;
  }
  __syncthreads();
  if (n < 32) {
    int b = n >> 2, j = n & 3;
    float L0 = bl[b][0], L1 = bl[b][1], L2 = bl[b][2], L3 = bl[b][3];
    float v;
    if (j == 0)      v = L0 - 0.5f * L2;
    else if (j == 1) v = L1 - 0.5f * L3;
    else if (j == 2) v = L0 + 0.5f * L2;
    else             v = L1 + 0.5f * L3;
    v = fminf(fmaxf(v, 0.0f), 1.0f);
    boxOut[b * 4 + j] = v;
  }
}

// ---------------- Stage 3: adaptive-max ROI pool all 3 levels -> feat (padded to 16 rows)
// feat is 16 x 24576; rows 8..15 zeroed so WMMA A-tiles read clean zeros.
__global__ void k_roi(const float* __restrict__ x1,
                      const float* __restrict__ x2,
                      const float* __restrict__ x3,
                      const float* __restrict__ box,
                      float* __restrict__ feat) {
  int idx = blockIdx.x * blockDim.x + threadIdx.x;   // 16*192*16*8 = 393216
  if (idx >= 16 * 192 * 16 * 8) return;
  int j  = idx & 7;
  int i  = (idx >> 3) & 15;
  int cc = (idx >> 7) % 192;
  int b  = idx / (128 * 192);
  int fi = b * DCLS + cc * 128 + i * 8 + j;
  if (b >= B_SZ) { feat[fi] = 0.0f; return; }

  const float* x; int H, W, c;
  if (cc < 64)       { x = x1; H = 128; W = 128; c = cc; }
  else if (cc < 128) { x = x2; H = 64;  W = 64;  c = cc - 64; }
  else               { x = x3; H = 32;  W = 32;  c = cc - 128; }

  float bn0 = box[b * 4 + 0], bn1 = box[b * 4 + 1];
  float bn2 = box[b * 4 + 2], bn3 = box[b * 4 + 3];
  int X1 = (int)floorf(bn0 * (float)W), Y1 = (int)floorf(bn1 * (float)H);
  int X2 = (int)floorf(bn2 * (float)W), Y2 = (int)floorf(bn3 * (float)H);
  bool valid = (X1 >= 0) && (Y1 >= 0) && (X2 < W) && (Y2 < H) && (X2 > X1) && (Y2 > Y1);
  int h = max(Y2 - Y1 + 1, 1);
  int w = max(X2 - X1 + 1, 1);

  int rlo = Y1 + (i * h) / 16;
  int rhi = Y1 + min(((i + 1) * h + 15) / 16, h);
  int clo = X1 + (j * w) / 8;
  int chi = X1 + min(((j + 1) * w + 7) / 8, w);
  int r0 = max(rlo, 0), r1 = min(rhi, H);
  int c0 = max(clo, 0), c1 = min(chi, W);

  float m = NEGV;
  const float* base = x + ((size_t)(b * C_SZ + c)) * H * W;
  for (int r = r0; r < r1; ++r) {
    const float* row = base + (size_t)r * W;
    for (int cx = c0; cx < c1; ++cx) m = fmaxf(m, row[cx]);
  }
  bool ok = valid && (m > -5e29f);
  feat[fi] = ok ? m : 0.0f;
}

// ---------------- Stage 4: WMMA head GEMM: feat(16x24576) x head_w1(7,24576,128) ---------
// One wave per (head, n-tile, k-chunk): 7*8*24 = 1344 waves, each K=1024 via 256 WMMAs.
__global__ void k_head_wmma(const float* __restrict__ feat,
                            const float* __restrict__ hw1,
                            float* __restrict__ part) {
  int wave = blockIdx.x * (blockDim.x >> 5) + (threadIdx.x >> 5);
  int lane = threadIdx.x & 31;
  int head   = wave / 192;
  int rest   = wave % 192;
  int ntile  = rest / 24;
  int kchunk = rest % 24;

  int m  = lane & 15;          // A row / B col index
  int kh = (lane >> 4) << 1;   // 0 for lanes 0-15, 2 for lanes 16-31

  const float* arow = feat + (size_t)m * DCLS + kchunk * 1024 + kh;
  const float* bcol = hw1 + (size_t)head * DCLS * 128
                          + ((size_t)(kchunk * 1024) + kh) * 128
                          + ntile * 16 + m;

  v8f acc = {0.f, 0.f, 0.f, 0.f, 0.f, 0.f, 0.f, 0.f};
  for (int kk = 0; kk < 1024; kk += 4) {
    v2f a = *(const v2f*)(arow + kk);        // (K=kh, K=kh+1) for this lane's row
    const float* bp = bcol + (size_t)kk * 128;
    v2f bv;
    bv.x = bp[0];                            // (K=kh,   N=m)
    bv.y = bp[128];                          // (K=kh+1, N=m)
    acc = __builtin_amdgcn_wmma_f32_16x16x4_f32(
        /*neg_a=*/false, a, /*neg_b=*/false, bv,
        /*c_mod=*/(short)0, acc, /*reuse_a=*/false, /*reuse_b=*/false);
  }
  *(v8f*)(part + (size_t)wave * 256 + lane * 8) = acc;
}

// ---------------- Stage 5: deterministic split-K reduce + bias + ReLU -> hidden(7,8,128) -
__global__ void k_head_reduce(const float* __restrict__ part,
                              const float* __restrict__ hb1,
                              float* __restrict__ hid) {
  int idx = blockIdx.x * blockDim.x + threadIdx.x;   // 7*8*128 = 7168
  if (idx >= 7168) return;
  int head = idx / 1024;
  int b    = (idx >> 7) & 7;
  int hcol = idx & 127;
  int ntile = hcol >> 4, nn = hcol & 15;
  // C/D layout: element (m=b<8, n=nn) lives at lane=nn, vgpr=b -> slot nn*8+b
  float s = hb1[head * 128 + hcol];
  int baseWave = (head * 8 + ntile) * 24;
  for (int kc = 0; kc < 24; ++kc)
    s += part[(size_t)(baseWave + kc) * 256 + nn * 8 + b];
  hid[idx] = fmaxf(s, 0.0f);
}

// ---------------- Stage 6: tiny output heads -> d_out (1904 floats, tuple order) ---------
__global__ void k_out(const float* __restrict__ hid,
                      const float* __restrict__ wp, const float* __restrict__ bp,
                      const float* __restrict__ wa, const float* __restrict__ ba,
                      const float* __restrict__ wad, const float* __restrict__ bad,
                      float* __restrict__ out) {
  int idx = blockIdx.x * blockDim.x + threadIdx.x;
  if (idx >= 1904) return;
  float s;
  if (idx < 304) {                 // o1: (8,38) from hidden[0]
    int b = idx / 38, o = idx % 38;
    s = bp[o];
    const float* hrow = hid + b * 128;
    for (int h = 0; h < 128; ++h) s += hrow[h] * wp[h * 38 + o];
  } else if (idx < 504) {          // o2: (8,25) from hidden[1]
    int t = idx - 304; int b = t / 25, o = t % 25;
    s = ba[o];
    const float* hrow = hid + 1024 + b * 128;
    for (int h = 0; h < 128; ++h) s += hrow[h] * wa[h * 25 + o];
  } else {                         // o_ad[a]: (8,35) from hidden[2+a]
    int t = idx - 504; int a = t / 280; int u = t % 280;
    int b = u / 35, o = u % 35;
    s = bad[a * 35 + o];
    const float* hrow = hid + (2 + a) * 1024 + b * 128;
    const float* wm = wad + a * 128 * 35;
    for (int h = 0; h < 128; ++h) s += hrow[h] * wm[h * 35 + o];
  }
  out[idx] = s;
}

extern "C" void kernel_launch(void* const* d_in, const int* in_sizes, int n_in,
                              void* d_out, int out_size, void* d_ws, size_t ws_size,
                              hipStream_t stream) {
  (void)in_sizes; (void)n_in; (void)out_size; (void)ws_size;
  const float* x1  = (const float*)d_in[0];
  const float* x2  = (const float*)d_in[1];
  const float* x3  = (const float*)d_in[2];
  // d_in[3] detection_targets, d_in[4] YI: unused by reference
  const float* w1  = (const float*)d_in[5];
  const float* b1  = (const float*)d_in[6];
  const float* w2  = (const float*)d_in[7];
  const float* b2  = (const float*)d_in[8];
  const float* w3  = (const float*)d_in[9];
  const float* b3  = (const float*)d_in[10];
  const float* hw1 = (const float*)d_in[11];
  const float* hb1 = (const float*)d_in[12];
  const float* wp  = (const float*)d_in[13];
  const float* bp  = (const float*)d_in[14];
  const float* wa  = (const float*)d_in[15];
  const float* ba  = (const float*)d_in[16];
  const float* wad = (const float*)d_in[17];
  const float* bad = (const float*)d_in[18];

  float* ws    = (float*)d_ws;
  float* part1 = ws;                 // 64*8*100            = 51200
  float* box   = ws + 51200;         // 8*4                 = 32
  float* feat  = ws + 51232;         // 16*24576            = 393216 (32B aligned)
  float* part2 = ws + 444448;        // 1344 waves * 256    = 344064 (32B aligned)
  float* hid   = ws + 788512;        // 7*8*128             = 7168
  // total ws: 795680 floats ~ 3.04 MB

  k_gemm1_partial<<<64, 128, 0, stream>>>(x3, w1, part1);
  k_box<<<1, 128, 0, stream>>>(part1, b1, w2, b2, w3, b3, box);
  k_roi<<<1536, 256, 0, stream>>>(x1, x2, x3, box, feat);
  k_head_wmma<<<168, 256, 0, stream>>>(feat, hw1, part2);
  k_head_reduce<<<28, 256, 0, stream>>>(part2, hb1, hid);
  k_out<<<8, 256, 0, stream>>>(hid, wp, ba == nullptr ? bp : bp, wa, ba, wad, bad, (float*)d_out);
}